// Label_Extract_51866025066826
// MI455X (gfx1250) — compile-verified
//
#include <hip/hip_runtime.h>

#define CFEAT 32

typedef __attribute__((ext_vector_type(2))) float v2f;
typedef __attribute__((ext_vector_type(8))) float v8f;

__device__ __forceinline__ void atomAddF(float* p, float v) {
  __hip_atomic_fetch_add(p, v, __ATOMIC_RELAXED, __HIP_MEMORY_SCOPE_AGENT);
}

// ---------------------------------------------------------------------------
// Per-edge degree counting (float counts so they can be used directly in rsqrt;
// degrees << 2^24 so float accumulation is exact).
// ---------------------------------------------------------------------------
__global__ __launch_bounds__(256) void deg_kernel(
    const int* __restrict__ row, const int* __restrict__ col,
    float* __restrict__ dr, float* __restrict__ dc, int E) {
  int e = blockIdx.x * 256 + threadIdx.x;
  if (e >= E) return;
  atomAddF(&dr[row[e]], 1.0f);
  atomAddF(&dc[col[e]], 1.0f);
}

// ---------------------------------------------------------------------------
// Per-edge weights w (directed; norm == norm1 algebraically) and wu
// (undirected), plus the self-return accumulators re_row/re_col/re_u = sum w^2.
// ---------------------------------------------------------------------------
__global__ __launch_bounds__(256) void weight_kernel(
    const int* __restrict__ row, const int* __restrict__ col,
    const float* __restrict__ dr, const float* __restrict__ dc,
    float* __restrict__ re_row, float* __restrict__ re_col,
    float* __restrict__ re_u,
    float* __restrict__ w, float* __restrict__ wu, int E) {
  int e = blockIdx.x * 256 + threadIdx.x;
  if (e >= E) return;
  int r = row[e], c = col[e];
  float drr = dr[r];          // deg_row[r]
  float dcc = dc[c];          // deg_col[c]
  float dur = drr + dc[r];    // undirected degree of r
  float duc = dr[c] + dcc;    // undirected degree of c
  float we  = rsqrtf(dcc + 1e-9f) * rsqrtf(drr + 1e-9f);
  float wue = rsqrtf(duc + 1e-9f) * rsqrtf(dur + 1e-9f);
  w[e]  = we;
  wu[e] = wue;
  float w2 = we * we, wu2 = wue * wue;
  atomAddF(&re_row[r], w2);
  atomAddF(&re_col[c], w2);
  atomAddF(&re_u[r], wu2);
  atomAddF(&re_u[c], wu2);
}

// ---------------------------------------------------------------------------
// Propagation pass 1 (everything that reads only x = label):
//   h1[r] += w*x[c];  h2[c] += w*x[r];  x1[c] += wu*x[r];  x1[r] += wu*x[c]
// One wave32 per edge, lane = feature (C == 32): coalesced 128B gathers and
// coalesced global_atomic_add_f32 scatter segments, all L2-resident.
// ---------------------------------------------------------------------------
__global__ __launch_bounds__(256) void prop1_kernel(
    const int* __restrict__ row, const int* __restrict__ col,
    const float* __restrict__ w, const float* __restrict__ wu,
    const float* __restrict__ x,
    float* __restrict__ h1, float* __restrict__ h2, float* __restrict__ x1,
    int E) {
  int t = blockIdx.x * 256 + threadIdx.x;
  int e = t >> 5;
  int lane = t & 31;
  if (e >= E) return;
  int r = row[e], c = col[e];
  float we = w[e], wue = wu[e];
  float xc = x[c * CFEAT + lane];
  float xr = x[r * CFEAT + lane];
  atomAddF(&h1[r * CFEAT + lane], we * xc);
  atomAddF(&h2[c * CFEAT + lane], we * xr);
  atomAddF(&x1[c * CFEAT + lane], wue * xr);
  atomAddF(&x1[r * CFEAT + lane], wue * xc);
}

// ---------------------------------------------------------------------------
// Propagation pass 2 (reads h1, h2, x1 produced by pass 1):
//   h3[r] += w*h2[c];  h4[c] += w*h1[r];  x2[c] += wu*x1[r];  x2[r] += wu*x1[c]
// ---------------------------------------------------------------------------
__global__ __launch_bounds__(256) void prop2_kernel(
    const int* __restrict__ row, const int* __restrict__ col,
    const float* __restrict__ w, const float* __restrict__ wu,
    const float* __restrict__ h1, const float* __restrict__ h2,
    const float* __restrict__ x1,
    float* __restrict__ h3, float* __restrict__ h4, float* __restrict__ x2,
    int E) {
  int t = blockIdx.x * 256 + threadIdx.x;
  int e = t >> 5;
  int lane = t & 31;
  if (e >= E) return;
  int r = row[e], c = col[e];
  float we = w[e], wue = wu[e];
  float h2c = h2[c * CFEAT + lane];
  float h1r = h1[r * CFEAT + lane];
  float x1r = x1[r * CFEAT + lane];
  float x1c = x1[c * CFEAT + lane];
  atomAddF(&h3[r * CFEAT + lane], we * h2c);
  atomAddF(&h4[c * CFEAT + lane], we * h1r);
  atomAddF(&x2[c * CFEAT + lane], wue * x1r);
  atomAddF(&x2[r * CFEAT + lane], wue * x1c);
}

// ---------------------------------------------------------------------------
// Finalize: per 16-node tile, build the six adjusted 16x32 blocks in LDS,
// compute their row sums as H(16x32) x ones(32x16) with 8 chained
// V_WMMA_F32_16X16X4_F32 ops (fp32-exact), then write the row-normalized
// N x 192 output. One full wave per block -> EXEC all-1s for WMMA.
//
// A-matrix 16x4 fp32 layout (ISA 7.12.2): lane L holds M = L%16;
//   VGPR0 = K = 2*(L>=16), VGPR1 = K+1.  B = all ones.
// C/D 16x16 fp32 layout: lane L = column L%16; acc[q] = row q + 8*(L>=16).
// With B all-ones every column equals the row sum.
// ---------------------------------------------------------------------------
__global__ __launch_bounds__(32) void finalize_kernel(
    const float* __restrict__ label,
    const float* __restrict__ re_row, const float* __restrict__ re_col,
    const float* __restrict__ re_u,
    const float* __restrict__ h1, const float* __restrict__ h2,
    const float* __restrict__ h3, const float* __restrict__ h4,
    const float* __restrict__ x1, const float* __restrict__ x2,
    float* __restrict__ out, int N) {
  __shared__ float tile[6][16 * 33];   // stride 33 -> no LDS bank conflicts
  __shared__ float sums[6][16];
  const int lane  = threadIdx.x;
  const int node0 = blockIdx.x * 16;

  // Stage adjusted values (order matches output concat):
  // 0: x2 - label*re_u, 1: x1, 2: h3 - label*re_row, 3: h2, 4: h1,
  // 5: h4 - label*re_col
  for (int r = 0; r < 16; ++r) {
    int node = node0 + r;
    float lab = 0.f, rr = 0.f, rc = 0.f, ru = 0.f;
    float vx2 = 0.f, vx1 = 0.f, vh3 = 0.f, vh2 = 0.f, vh1 = 0.f, vh4 = 0.f;
    if (node < N) {
      int i = node * CFEAT + lane;
      lab = label[i];
      rr = re_row[node]; rc = re_col[node]; ru = re_u[node];
      vx2 = x2[i]; vx1 = x1[i]; vh3 = h3[i];
      vh2 = h2[i]; vh1 = h1[i]; vh4 = h4[i];
    }
    tile[0][r * 33 + lane] = vx2 - lab * ru;
    tile[1][r * 33 + lane] = vx1;
    tile[2][r * 33 + lane] = vh3 - lab * rr;
    tile[3][r * 33 + lane] = vh2;
    tile[4][r * 33 + lane] = vh1;
    tile[5][r * 33 + lane] = vh4 - lab * rc;
  }
  __syncthreads();

  const int m  = lane & 15;
  const int hi = lane >> 4;
  v2f b; b.x = 1.0f; b.y = 1.0f;     // ones(4x16) fragment

  for (int blk = 0; blk < 6; ++blk) {
    v8f acc = {};
#pragma unroll
    for (int kb = 0; kb < CFEAT; kb += 4) {
      v2f a;
      a.x = tile[blk][m * 33 + kb + 2 * hi];
      a.y = tile[blk][m * 33 + kb + 2 * hi + 1];
      acc = __builtin_amdgcn_wmma_f32_16x16x4_f32(
          false, a, false, b, (short)0, acc, false, false);
    }
    if (m == 0) {
#pragma unroll
      for (int q = 0; q < 8; ++q) sums[blk][hi * 8 + q] = acc[q];
    }
  }
  __syncthreads();

  for (int r = 0; r < 16; ++r) {
    int node = node0 + r;
    if (node >= N) break;
    int o = node * (6 * CFEAT);
#pragma unroll
    for (int blk = 0; blk < 6; ++blk)
      out[o + blk * CFEAT + lane] =
          tile[blk][r * 33 + lane] / (sums[blk][r] + 1e-5f);
  }
}

// ---------------------------------------------------------------------------
// Host launcher. Inputs: d_in[0] = label (N*32 fp32), d_in[1] = edge_index
// (2*E int32, first E = row, next E = col). Output: N*192 fp32.
// Workspace layout (floats):
//   [0,N) deg_row | [N,2N) deg_col | [2N,3N) re_row | [3N,4N) re_col
//   [4N,5N) re_u  | [5N,5N+E) w    | [5N+E,5N+2E) wu | then h1,h2,h3,h4,x1,x2
// Total = (197*N + 2*E)*4 bytes  (~92 MB here).
// ---------------------------------------------------------------------------
extern "C" void kernel_launch(void* const* d_in, const int* in_sizes, int n_in,
                              void* d_out, int out_size, void* d_ws,
                              size_t ws_size, hipStream_t stream) {
  const float* label = (const float*)d_in[0];
  const int*   ei    = (const int*)d_in[1];
  const int N = in_sizes[0] / CFEAT;
  const int E = in_sizes[1] / 2;
  const int* row = ei;
  const int* col = ei + E;

  float* ws = (float*)d_ws;
  float* deg_row = ws;
  float* deg_col = ws + (size_t)N;
  float* re_row  = ws + 2 * (size_t)N;
  float* re_col  = ws + 3 * (size_t)N;
  float* re_u    = ws + 4 * (size_t)N;
  float* w       = ws + 5 * (size_t)N;
  float* wu      = w + (size_t)E;
  float* h1      = wu + (size_t)E;
  float* h2      = h1 + (size_t)N * CFEAT;
  float* h3      = h2 + (size_t)N * CFEAT;
  float* h4      = h3 + (size_t)N * CFEAT;
  float* x1      = h4 + (size_t)N * CFEAT;
  float* x2      = x1 + (size_t)N * CFEAT;

  size_t zero_bytes = (5 * (size_t)N + 2 * (size_t)E +
                       6 * (size_t)N * CFEAT) * sizeof(float);
  hipMemsetAsync(d_ws, 0, zero_bytes, stream);

  const int tpb = 256;
  int eb  = (E + tpb - 1) / tpb;       // 1 thread / edge
  int ewb = (E + 7) / 8;               // 1 wave32 / edge, 8 waves / block

  deg_kernel<<<eb, tpb, 0, stream>>>(row, col, deg_row, deg_col, E);
  weight_kernel<<<eb, tpb, 0, stream>>>(row, col, deg_row, deg_col,
                                        re_row, re_col, re_u, w, wu, E);
  prop1_kernel<<<ewb, tpb, 0, stream>>>(row, col, w, wu, label,
                                        h1, h2, x1, E);
  prop2_kernel<<<ewb, tpb, 0, stream>>>(row, col, w, wu, h1, h2, x1,
                                        h3, h4, x2, E);
  finalize_kernel<<<(N + 15) / 16, 32, 0, stream>>>(
      label, re_row, re_col, re_u, h1, h2, h3, h4, x1, x2,
      (float*)d_out, N);
}